// MoEGate_65807488909558
// MI455X (gfx1250) — compile-verified
//
#include <hip/hip_runtime.h>
#include <math.h>

typedef float v2f __attribute__((ext_vector_type(2)));
typedef float v8f __attribute__((ext_vector_type(8)));

#define T_TOK 16384
#define HDIM  2048
#define NEXP  64
#define NBATCH 4
#define SEQ   4096
#define ALPHA 0.1f
#define EPSV  1e-20f

__global__ void moe_gate_zero_ws(float* __restrict__ ws) {
    int i = blockIdx.x * blockDim.x + threadIdx.x;
    if (i < 2 * NBATCH * NEXP) ws[i] = 0.0f;
}

// One block = 256 threads = 8 waves; each wave: 16 tokens x 64 experts via
// V_WMMA_F32_16X16X4_F32 (4 N-tiles, K-loop over 2048 in steps of 4).
__global__ __launch_bounds__(256) void moe_gate_main(
    const float* __restrict__ x,      // [16384, 2048]
    const float* __restrict__ w,      // [64, 2048]
    float* __restrict__ out,          // [65537] : idx(32768) | weight(32768) | aux(1)
    float* __restrict__ acc_ws)       // [2][4][64] : score sums, then counts
{
    __shared__ float tile[8][16][68];   // per-wave logits tile, padded
    __shared__ float wgScore[NEXP];
    __shared__ float wgCount[NEXP];

    const int tid    = threadIdx.x;
    const int wave   = tid >> 5;
    const int lane   = tid & 31;
    const int laneLo = lane & 15;
    const int laneHi = lane >> 4;       // 0: K 0..1 / rows 0..7 ; 1: K 2..3 / rows 8..15

    if (tid < NEXP) { wgScore[tid] = 0.0f; wgCount[tid] = 0.0f; }

    const int tok0 = blockIdx.x * 128 + wave * 16;
    const int b    = (blockIdx.x * 128) >> 12;   // 4096 tokens per batch

    // A fragment: row = tok0+laneLo, elements k+2*laneHi .. +1  (global_load_b64)
    const float* xptr = x + (size_t)(tok0 + laneLo) * HDIM + 2 * laneHi;
    // B fragment for N-tile t: expert = 16*t + laneLo, same K pattern
    const float* wptr = w + (size_t)laneLo * HDIM + 2 * laneHi;

    v8f acc0 = {}, acc1 = {}, acc2 = {}, acc3 = {};

    #pragma unroll 4
    for (int k = 0; k < HDIM; k += 4) {
        v2f a  = *(const v2f*)(xptr + k);
        v2f b0 = *(const v2f*)(wptr + 0 * 16 * HDIM + k);
        v2f b1 = *(const v2f*)(wptr + 1 * 16 * HDIM + k);
        v2f b2 = *(const v2f*)(wptr + 2 * 16 * HDIM + k);
        v2f b3 = *(const v2f*)(wptr + 3 * 16 * HDIM + k);
        acc0 = __builtin_amdgcn_wmma_f32_16x16x4_f32(false, a, false, b0, (short)0, acc0, false, false);
        acc1 = __builtin_amdgcn_wmma_f32_16x16x4_f32(false, a, false, b1, (short)0, acc1, false, false);
        acc2 = __builtin_amdgcn_wmma_f32_16x16x4_f32(false, a, false, b2, (short)0, acc2, false, false);
        acc3 = __builtin_amdgcn_wmma_f32_16x16x4_f32(false, a, false, b3, (short)0, acc3, false, false);
    }

    // C/D layout: VGPR r, lanes 0-15 -> M=r, lanes 16-31 -> M=r+8; N = 16*t + laneLo
    #pragma unroll
    for (int r = 0; r < 8; ++r) {
        int row = r + 8 * laneHi;
        tile[wave][row][ 0 + laneLo] = acc0[r];
        tile[wave][row][16 + laneLo] = acc1[r];
        tile[wave][row][32 + laneLo] = acc2[r];
        tile[wave][row][48 + laneLo] = acc3[r];
    }

    __syncthreads();

    if (laneHi == 0) {
        const int   row = laneLo;
        const int   tok = tok0 + row;
        const float* l  = tile[wave][row];

        // single-pass top-2 (strict > keeps lowest index on ties, like lax.top_k)
        float m1 = -INFINITY, m2 = -INFINITY;
        int   i1 = 0, i2 = 0;
        for (int e = 0; e < NEXP; ++e) {
            float v = l[e];
            if (v > m1)      { m2 = m1; i2 = i1; m1 = v; i1 = e; }
            else if (v > m2) { m2 = v; i2 = e; }
        }
        float sum = 0.0f;
        for (int e = 0; e < NEXP; ++e) sum += __expf(l[e] - m1);
        float inv = 1.0f / sum;

        // per-expert softmax score sums for aux loss (staggered: bank-conflict free)
        for (int j = 0; j < NEXP; ++j) {
            int e = (row + j) & (NEXP - 1);
            atomicAdd(&wgScore[e], __expf(l[e] - m1) * inv);
        }
        atomicAdd(&wgCount[i1], 1.0f);
        atomicAdd(&wgCount[i2], 1.0f);

        float w1 = inv;                     // exp(m1-m1)*inv
        float w2 = __expf(m2 - m1) * inv;
        float dn = 1.0f / (w1 + w2 + EPSV);
        out[2 * tok + 0] = (float)i1;
        out[2 * tok + 1] = (float)i2;
        out[2 * T_TOK + 2 * tok + 0] = w1 * dn;
        out[2 * T_TOK + 2 * tok + 1] = w2 * dn;
    }

    __syncthreads();

    if (tid < NEXP) {
        atomicAdd(&acc_ws[b * NEXP + tid],                 wgScore[tid]);
        atomicAdd(&acc_ws[NBATCH * NEXP + b * NEXP + tid], wgCount[tid]);
    }
}

__global__ void moe_gate_finalize(const float* __restrict__ acc_ws,
                                  float* __restrict__ out) {
    __shared__ float red[NEXP];
    const int e = threadIdx.x;            // 64 threads
    float p = 0.0f;
    for (int b = 0; b < NBATCH; ++b) {
        float ce = acc_ws[NBATCH * NEXP + b * NEXP + e] * ((float)NEXP / (SEQ * 2));
        float es = acc_ws[b * NEXP + e] * (1.0f / (float)SEQ);
        p += ce * es;
    }
    red[e] = p;
    __syncthreads();
    for (int s = 32; s > 0; s >>= 1) {
        if (e < s) red[e] += red[e + s];
        __syncthreads();
    }
    if (e == 0) out[4 * T_TOK] = ALPHA * red[0] / (float)NBATCH;
}

extern "C" void kernel_launch(void* const* d_in, const int* in_sizes, int n_in,
                              void* d_out, int out_size, void* d_ws, size_t ws_size,
                              hipStream_t stream) {
    const float* x  = (const float*)d_in[0];   // hidden_states [4,4096,2048] f32
    const float* w  = (const float*)d_in[1];   // weight [64,2048] f32
    float* out = (float*)d_out;
    float* ws  = (float*)d_ws;                 // 512 floats used

    moe_gate_zero_ws<<<1, 512, 0, stream>>>(ws);
    moe_gate_main<<<T_TOK / 128, 256, 0, stream>>>(x, w, out, ws);
    moe_gate_finalize<<<1, NEXP, 0, stream>>>(ws, out);
}